// SymplecticGating_4346506904317
// MI455X (gfx1250) — compile-verified
//
#include <hip/hip_runtime.h>
#include <math.h>

#define DIM   1024
#define ROWS  16384   // B*S = 4*4096
#define SEQ   4096
#define TOPK  64
#define LP    1028    // padded LDS row stride (floats) -> conflict-free A reads

typedef __attribute__((ext_vector_type(2))) float v2f;
typedef __attribute__((ext_vector_type(8))) float v8f;

__device__ __forceinline__ int wave_sum_i(int v) {
#pragma unroll
  for (int o = 16; o > 0; o >>= 1) v += __shfl_xor(v, o, 32);
  return v;
}
__device__ __forceinline__ float wave_sum_f(float v) {
#pragma unroll
  for (int o = 16; o > 0; o >>= 1) v += __shfl_xor(v, o, 32);
  return v;
}

// Async global->LDS stage of a 16x1024 f32 tile into a padded [16][LP] buffer.
// Each of 512 threads copies 8 aligned 16-byte chunks via
// GLOBAL_LOAD_ASYNC_TO_LDS_B128 (ASYNCcnt); caller must __syncthreads() after.
__device__ __forceinline__ void stage_tile_async(const float* __restrict__ gbase,
                                                 float* lds, int tid) {
#pragma unroll
  for (int j = 0; j < 8; ++j) {
    int idx4 = (tid + j * 512) * 4;          // float index within 16x1024 tile
    int r = idx4 >> 10;
    int c = idx4 & (DIM - 1);
    const float* g = gbase + (size_t)r * DIM + c;
    unsigned ldsoff = (unsigned)(uintptr_t)(&lds[r * LP + c]); // low 32b = LDS offset
    asm volatile("global_load_async_to_lds_b128 %0, %1, off"
                 :: "v"(ldsoff), "v"(g) : "memory");
  }
  asm volatile("s_wait_asynccnt 0x0" ::: "memory");
}

// ---------------------------------------------------------------------------
// K1: gate_pre & mag_pre GEMMs (WMMA f32) + per-row top-64 + masked xg
// block = 512 threads (16 waves), one block per 16 rows
// ---------------------------------------------------------------------------
__global__ __launch_bounds__(512)
void k1_gate_mag_topk(const float* __restrict__ x,
                      const float* __restrict__ gate_W,
                      const float* __restrict__ gate_b,
                      const float* __restrict__ mag_W,
                      const float* __restrict__ mag_b,
                      float* __restrict__ xg) {
  __shared__ float xs[16 * LP];   // x tile during GEMM, then mag_pre
  __shared__ float gp[16 * LP];   // gate_pre
  const int row0 = blockIdx.x * 16;
  const int tid  = threadIdx.x;

  stage_tile_async(x + (size_t)row0 * DIM, xs, tid);
  __syncthreads();

  const int wave = tid >> 5, lane = tid & 31;
  const int n0   = wave * 64;
  const int m    = lane & 15;          // row (A) / col-in-tile (B,C)
  const int koff = (lane >> 4) * 2;    // K split across lane halves

  const v8f zero = {0.f,0.f,0.f,0.f,0.f,0.f,0.f,0.f};
  v8f cg[4], cm[4];
#pragma unroll
  for (int t = 0; t < 4; ++t) { cg[t] = zero; cm[t] = zero; }

  for (int k = 0; k < DIM; k += 4) {
    v2f a = *reinterpret_cast<const v2f*>(&xs[m * LP + k + koff]);
#pragma unroll
    for (int t = 0; t < 4; ++t) {
      int n = n0 + t * 16 + m;
      v2f bg = *reinterpret_cast<const v2f*>(&gate_W[(size_t)n * DIM + k + koff]);
      cg[t] = __builtin_amdgcn_wmma_f32_16x16x4_f32(false, a, false, bg,
                                                    (short)0, cg[t], false, false);
      v2f bm = *reinterpret_cast<const v2f*>(&mag_W[(size_t)n * DIM + k + koff]);
      cm[t] = __builtin_amdgcn_wmma_f32_16x16x4_f32(false, a, false, bm,
                                                    (short)0, cm[t], false, false);
    }
  }
  __syncthreads();   // done reading xs; safe to overwrite with mag_pre

  const int rbase = (lane >> 4) * 8;
#pragma unroll
  for (int t = 0; t < 4; ++t) {
    int col = n0 + t * 16 + m;
#pragma unroll
    for (int v = 0; v < 8; ++v) {
      int row = v + rbase;
      gp[row * LP + col] = cg[t][v] + gate_b[col];
      xs[row * LP + col] = cm[t][v] + mag_b[col];
    }
  }
  __syncthreads();

  // ---- top-64 per row: wave r handles row r ----
  const int r = wave;
  unsigned key[32];
#pragma unroll
  for (int i = 0; i < 32; ++i) {
    unsigned u = __float_as_uint(gp[r * LP + i * 32 + lane]);
    key[i] = (u & 0x80000000u) ? ~u : (u | 0x80000000u);  // order-preserving
  }
  unsigned pivot = 0;
  for (int bit = 31; bit >= 0; --bit) {
    unsigned trial = pivot | (1u << bit);
    int local = 0;
#pragma unroll
    for (int i = 0; i < 32; ++i) local += (key[i] >= trial);
    if (wave_sum_i(local) >= TOPK) pivot = trial;   // pivot -> 64th largest
  }
  int local_gt = 0;
#pragma unroll
  for (int i = 0; i < 32; ++i) local_gt += (key[i] > pivot);
  const int quota = TOPK - wave_sum_i(local_gt);    // ties to accept (by index)
  const unsigned below = (1u << lane) - 1u;

  int running = 0;
  for (int i = 0; i < 32; ++i) {
    bool eq = (key[i] == pivot);
    unsigned bal = (unsigned)__ballot(eq);
    bool take   = eq && (running + __popc(bal & below) < quota);
    bool accept = (key[i] > pivot) || take;
    running += __popc(bal);
    int col = i * 32 + lane;
    float gpre = gp[r * LP + col];
    float mpre = xs[r * LP + col];
    float val  = (accept && gpre > 0.f) ? fmaxf(mpre, 0.f) : 0.f;
    xg[(size_t)(row0 + r) * DIM + col] = val;
  }
}

// ---------------------------------------------------------------------------
// K2: q / k / phase GEMMs (WMMA f32) + L2 normalization
// grid = (ROWS/16, 3); blockIdx.y: 0 -> q, 1 -> k, 2 -> phase
// ---------------------------------------------------------------------------
__global__ __launch_bounds__(512)
void k2_qkp(const float* __restrict__ xg,
            const float* __restrict__ Wq,
            const float* __restrict__ Wk,
            const float* __restrict__ Wp,
            float* __restrict__ qo,
            float* __restrict__ ko,
            float* __restrict__ po) {
  __shared__ float xs[16 * LP];
  const int mode = blockIdx.y;
  const float* __restrict__ W = (mode == 0) ? Wq : (mode == 1) ? Wk : Wp;
  float* __restrict__ out     = (mode == 0) ? qo : (mode == 1) ? ko : po;

  const int row0 = blockIdx.x * 16;
  const int tid  = threadIdx.x;

  stage_tile_async(xg + (size_t)row0 * DIM, xs, tid);
  __syncthreads();

  const int wave = tid >> 5, lane = tid & 31;
  const int n0 = wave * 64, m = lane & 15, koff = (lane >> 4) * 2;
  const v8f zero = {0.f,0.f,0.f,0.f,0.f,0.f,0.f,0.f};
  v8f c[4];
#pragma unroll
  for (int t = 0; t < 4; ++t) c[t] = zero;

  for (int k = 0; k < DIM; k += 4) {
    v2f a = *reinterpret_cast<const v2f*>(&xs[m * LP + k + koff]);
#pragma unroll
    for (int t = 0; t < 4; ++t) {
      int n = n0 + t * 16 + m;
      v2f b = *reinterpret_cast<const v2f*>(&W[(size_t)n * DIM + k + koff]);
      c[t] = __builtin_amdgcn_wmma_f32_16x16x4_f32(false, a, false, b,
                                                   (short)0, c[t], false, false);
    }
  }
  __syncthreads();   // xs free -> store pre-activation

  const int rbase = (lane >> 4) * 8;
#pragma unroll
  for (int t = 0; t < 4; ++t) {
    int col = n0 + t * 16 + m;
#pragma unroll
    for (int v = 0; v < 8; ++v) xs[(v + rbase) * LP + col] = c[t][v];
  }
  __syncthreads();

  if (mode < 2) {            // row-wise L2 norm: wave r owns row r
    const int r = wave;
    float s = 0.f;
#pragma unroll
    for (int i = 0; i < 32; ++i) { float v = xs[r * LP + i * 32 + lane]; s += v * v; }
    s = wave_sum_f(s);
    float scale = 1.f / fmaxf(sqrtf(s), 1e-12f);
#pragma unroll
    for (int i = 0; i < 32; ++i) {
      int col = i * 32 + lane;
      out[(size_t)(row0 + r) * DIM + col] = xs[r * LP + col] * scale;
    }
  } else {                   // pair-wise L2 norm over (2p, 2p+1)
    for (int i = tid; i < 16 * DIM; i += 512) {
      int r = i >> 10, col = i & (DIM - 1);
      float v = xs[r * LP + col];
      float w = xs[r * LP + (col ^ 1)];
      float nn = sqrtf(v * v + w * w);
      out[(size_t)(row0 + r) * DIM + col] = v / fmaxf(nn, 1e-12f);
    }
  }
}

// ---------------------------------------------------------------------------
// K3: momentum / phase-delta combine. One block per (b, n) output element.
// ---------------------------------------------------------------------------
__global__ __launch_bounds__(256)
void k3_combine(const float* __restrict__ q,
                const float* __restrict__ kk,
                const float* __restrict__ ph,
                float* __restrict__ out) {
  __shared__ float red[256];
  const int r = blockIdx.x;
  const int n = r & (SEQ - 1);
  const int tid = threadIdx.x;
  if (n == 0) { if (tid == 0) out[r] = 0.f; return; }

  const float* qc = q  + (size_t)r * DIM;
  const float* qp = q  + (size_t)(r - 1) * DIM;
  const float* kc = kk + (size_t)r * DIM;
  const float* kp = kk + (size_t)(r - 1) * DIM;
  const float* pc = ph + (size_t)r * DIM;
  const float* pp = ph + (size_t)(r - 1) * DIM;

  float mm = 0.f;
  for (int d = tid; d < DIM; d += 256) {
    float t = qc[d] * kp[d] - qp[d] * kc[d];
    mm += t * t;
  }
  float pd = 0.f;
  for (int p = tid; p < DIM / 2; p += 256) {
    float a0 = pp[2 * p], a1 = pp[2 * p + 1];
    float b0 = pc[2 * p], b1 = pc[2 * p + 1];
    float cross = a0 * b1 - a1 * b0;
    float dot   = fminf(fmaxf(a0 * b0 + a1 * b1, -1.f), 1.f);
    pd += fabsf(atan2f(cross, dot));
  }
  red[tid] = mm; __syncthreads();
  for (int o = 128; o > 0; o >>= 1) { if (tid < o) red[tid] += red[tid + o]; __syncthreads(); }
  float mm_s = red[0]; __syncthreads();
  red[tid] = pd; __syncthreads();
  for (int o = 128; o > 0; o >>= 1) { if (tid < o) red[tid] += red[tid + o]; __syncthreads(); }
  if (tid == 0) {
    float comp = tanhf(sqrtf(mm_s));
    float ps   = red[0] / (512.0f * 3.14159265358979323846f);
    out[r] = 0.5f * comp + 0.5f * ps;
  }
}

// ---------------------------------------------------------------------------
extern "C" void kernel_launch(void* const* d_in, const int* in_sizes, int n_in,
                              void* d_out, int out_size, void* d_ws, size_t ws_size,
                              hipStream_t stream) {
  (void)in_sizes; (void)n_in; (void)out_size; (void)ws_size;
  const float* x       = (const float*)d_in[0];
  const float* gate_W  = (const float*)d_in[1];
  const float* gate_b  = (const float*)d_in[2];
  const float* mag_W   = (const float*)d_in[3];
  const float* mag_b   = (const float*)d_in[4];
  const float* Wq      = (const float*)d_in[5];
  const float* Wk      = (const float*)d_in[6];
  const float* phase_W = (const float*)d_in[7];
  float* out = (float*)d_out;

  const size_t RD = (size_t)ROWS * DIM;
  float* ws = (float*)d_ws;
  float* xg = ws;
  float* q  = ws + RD;
  float* kk = ws + 2 * RD;
  float* ph = ws + 3 * RD;

  k1_gate_mag_topk<<<dim3(ROWS / 16), dim3(512), 0, stream>>>(
      x, gate_W, gate_b, mag_W, mag_b, xg);
  k2_qkp<<<dim3(ROWS / 16, 3), dim3(512), 0, stream>>>(
      xg, Wq, Wk, phase_W, q, kk, ph);
  k3_combine<<<dim3(ROWS), dim3(256), 0, stream>>>(q, kk, ph, out);
}